// get_edge_featureori_13237089206321
// MI455X (gfx1250) — compile-verified
//
#include <hip/hip_runtime.h>

typedef __attribute__((ext_vector_type(2))) float v2f;
typedef __attribute__((ext_vector_type(8))) float v8f;

#define N_PTS   4096
#define BATCH   8
#define KNN     16
#define THREADS 128               // 4 wave32s
#define WAVES   (THREADS / 32)
#define ROWS_PER_WAVE 32
#define ROWS_PER_WG   (WAVES * ROWS_PER_WAVE)   // 128
#define TILES   (N_PTS / 16)      // 256 column tiles
#define CH      ((size_t)KNN * N_PTS)           // 65536 elems per channel slice

__launch_bounds__(THREADS)
__global__ void knn_edge_kernel(const float* __restrict__ pc,
                                float* __restrict__ ef,
                                int*   __restrict__ idxo)
{
    __shared__ float xs[N_PTS], ys[N_PTS], zs[N_PTS], sq[N_PTS];   // 64 KB
    __shared__ float tile[WAVES][ROWS_PER_WAVE][16];               //  8 KB

    const int tid = threadIdx.x;
    const int b   = blockIdx.x >> 5;            // 32 row-blocks per batch
    const int rb  = blockIdx.x & 31;
    const int wgRowBase = rb * ROWS_PER_WG;

    const float* px = pc + (size_t)b * 3 * N_PTS;
    const float* py = px + N_PTS;
    const float* pz = py + N_PTS;

    // Stage whole batch into LDS as SoA + precomputed squared norms.
    for (int i = tid; i < N_PTS; i += THREADS) {
        float x = px[i], y = py[i], z = pz[i];
        xs[i] = x; ys[i] = y; zs[i] = z;
        sq[i] = x * x + y * y + z * z;
    }
    __syncthreads();

    const int  wave = tid >> 5;
    const int  lane = tid & 31;
    const int  lm   = lane & 15;
    const bool hiH  = lane >= 16;
    const int  waveRowBase = wgRowBase + wave * ROWS_PER_WAVE;

    // Branchless per-lane helpers (loop-invariant).
    const float  yMask  = hiH ? 0.0f : 1.0f;     // zeroes K=3 row of B / A
    const float* bxBase = hiH ? zs : xs;         // K=2 comes from z for hi half

    // A tiles (16x4, K padded with 0), pre-scaled by -2.
    // Layout: lanes 0-15 hold K=0,1 (V0,V1); lanes 16-31 hold K=2,3.
    const int m0 = waveRowBase + lm;
    const int m1 = m0 + 16;
    v2f a0, a1;
    a0.x = -2.0f * bxBase[m0];
    a0.y = -2.0f * ys[m0] * yMask;
    a1.x = -2.0f * bxBase[m1];
    a1.y = -2.0f * ys[m1] * yMask;

    // Row norms |row|^2 folded into the WMMA accumulator input C:
    // C/D layout: VGPR r holds row (rBase + r), lane's column fixed -> loop-invariant.
    v8f cAcc0, cAcc1;
    #pragma unroll
    for (int r = 0; r < 8; ++r) {
        int rr = waveRowBase + (hiH ? 8 : 0) + r;
        cAcc0[r] = sq[rr];
        cAcc1[r] = sq[rr + 16];
    }

    const int myRow = waveRowBase + lane;   // query row this lane selects for

    float bestD[KNN]; int bestI[KNN];
    #pragma unroll
    for (int i = 0; i < KNN; ++i) { bestD[i] = 3.4e38f; bestI[i] = 0; }
    float worstVal = 3.4e38f; int worstPos = 0;

    for (int j = 0; j < TILES; ++j) {
        const int n = (j << 4) + lm;
        // B tile (4x16): lanes 0-15 hold K=0,1 ; lanes 16-31 hold K=2,3.
        // Unconditional loads + multiplicative mask: no EXEC divergence near WMMA.
        v2f bb;
        bb.x = bxBase[n];
        bb.y = ys[n] * yMask;
        const float sqc = sq[n];            // D-layout: lane's column is fixed

        // D = (-2*A)*B + |row|^2   (accumulate via C operand)
        v8f d0 = __builtin_amdgcn_wmma_f32_16x16x4_f32(false, a0, false, bb,
                                                       (short)0, cAcc0, false, false);
        v8f d1 = __builtin_amdgcn_wmma_f32_16x16x4_f32(false, a1, false, bb,
                                                       (short)0, cAcc1, false, false);

        // d2 = D + |col|^2  (sqc is a single per-lane scalar)
        #pragma unroll
        for (int r = 0; r < 8; ++r) {
            d0[r] += sqc;
            d1[r] += sqc;
        }

        // Transpose D tiles through per-wave LDS scratch -> row per lane.
        const int rBase = hiH ? 8 : 0;
        #pragma unroll
        for (int r = 0; r < 8; ++r) {
            tile[wave][rBase + r][lm]      = d0[r];
            tile[wave][16 + rBase + r][lm] = d1[r];
        }
        asm volatile("s_wait_dscnt 0" ::: "memory");   // wave-internal LDS transpose

        float cand[16];
        #pragma unroll
        for (int q = 0; q < 16; ++q) cand[q] = tile[wave][lane][q];

        // Top-16 maintenance: unsorted array + tracked worst (all static indices).
        #pragma unroll
        for (int q = 0; q < 16; ++q) {
            const float d   = cand[q];
            const int   col = (j << 4) + q;
            if (col != myRow && d < worstVal) {
                #pragma unroll
                for (int i = 0; i < KNN; ++i)
                    if (i == worstPos) { bestD[i] = d; bestI[i] = col; }
                worstVal = bestD[0]; worstPos = 0;
                #pragma unroll
                for (int i = 1; i < KNN; ++i)
                    if (bestD[i] > worstVal) { worstVal = bestD[i]; worstPos = i; }
            }
        }
    }

    // Sort ascending by distance (odd-even transposition network, static indices).
    #pragma unroll
    for (int ph = 0; ph < KNN; ++ph) {
        #pragma unroll
        for (int i = (ph & 1); i + 1 < KNN; i += 2) {
            if (bestD[i] > bestD[i + 1]) {
                float td = bestD[i]; bestD[i] = bestD[i + 1]; bestD[i + 1] = td;
                int   ti = bestI[i]; bestI[i] = bestI[i + 1]; bestI[i + 1] = ti;
            }
        }
    }

    // Emit edge features + indices. Lane owns column myRow -> coalesced stores.
    const float cx = xs[myRow], cy = ys[myRow], cz = zs[myRow];
    const size_t efB = (size_t)b * 6 * CH;
    const size_t ioB = (size_t)b * CH;

    #pragma unroll
    for (int kk = 0; kk < KNN; ++kk) {
        const int nb = bestI[kk];
        idxo[ioB + (size_t)kk * N_PTS + myRow] = nb;

        const float nx = xs[nb], ny = ys[nb], nz = zs[nb];
        const size_t base = efB + (size_t)kk * N_PTS + myRow;
        ef[base + 0 * CH] = cx;
        ef[base + 1 * CH] = cy;
        ef[base + 2 * CH] = cz;
        ef[base + 3 * CH] = nx - cx;
        ef[base + 4 * CH] = ny - cy;
        ef[base + 5 * CH] = nz - cz;
    }
}

extern "C" void kernel_launch(void* const* d_in, const int* in_sizes, int n_in,
                              void* d_out, int out_size, void* d_ws, size_t ws_size,
                              hipStream_t stream)
{
    (void)in_sizes; (void)n_in; (void)d_ws; (void)ws_size; (void)out_size;

    const float* pc = (const float*)d_in[0];       // (8,3,4096) f32; d_in[1] = k (always 16)
    float* ef  = (float*)d_out;                    // (8,6,16,4096) f32
    int*   idx = (int*)(ef + (size_t)BATCH * 6 * CH); // (8,16,4096) i32, concatenated

    dim3 grid(BATCH * (N_PTS / ROWS_PER_WG));      // 8 * 32 = 256 blocks
    dim3 block(THREADS);
    knn_edge_kernel<<<grid, block, 0, stream>>>(pc, ef, idx);
}